// GCNConv_dgl_attn_28192165331245
// MI455X (gfx1250) — compile-verified
//
#include <hip/hip_runtime.h>

typedef float v2f __attribute__((ext_vector_type(2)));
typedef float v8f __attribute__((ext_vector_type(8)));

#define N_NODES 50000
#define IN_DIM  512
#define OUT_DIM 256
#define N_EDGES 800000

// ---------------------------------------------------------------------------
// Kernel 0: zero-init the output (segment_sum baseline; harness poisons d_out)
// ---------------------------------------------------------------------------
__global__ void zero_out_kernel(float4* __restrict__ out, int n4) {
    int i = blockIdx.x * blockDim.x + threadIdx.x;
    if (i < n4) out[i] = make_float4(0.f, 0.f, 0.f, 0.f);
}

// ---------------------------------------------------------------------------
// Kernel 1: h = x @ W^T + b  via V_WMMA_F32_16X16X4_F32 (fp32-exact path)
//
// One wave computes a 16(M) x 64(N) tile of h: 4 accumulators, A fragment
// reused across the 4 N-subtiles each K-step. A/B fragments are contiguous
// float2 loads per the ISA 32-bit 16x4 layout:
//   lane<16 : {K+0,K+1} of row (lane)     lane>=16 : {K+2,K+3} of row (lane-16)
// ---------------------------------------------------------------------------
__global__ void gemm_wmma_f32_kernel(const float* __restrict__ x,
                                     const float* __restrict__ W,
                                     const float* __restrict__ bias,
                                     float* __restrict__ h) {
    const int wave = threadIdx.x >> 5;                  // 0..7 (8 waves/block)
    const int lane = threadIdx.x & 31;
    const int tile = blockIdx.x * 8 + wave;             // global wave id
    const int m0 = (tile >> 2) * 16;                    // 3125 M tiles
    const int n0 = (tile & 3)  * 64;                    // 4 N groups of 64
    if (m0 >= N_NODES) return;                          // wave-uniform guard

    const int row   = lane & 15;                        // M row (A) / N col (B)
    const int khalf = (lane >> 4) << 1;                 // 0 or 2

    const float* xa = x + (size_t)(m0 + row) * IN_DIM + khalf;
    const float* w0 = W + (size_t)(n0 + row +  0) * IN_DIM + khalf;
    const float* w1 = W + (size_t)(n0 + row + 16) * IN_DIM + khalf;
    const float* w2 = W + (size_t)(n0 + row + 32) * IN_DIM + khalf;
    const float* w3 = W + (size_t)(n0 + row + 48) * IN_DIM + khalf;

    v8f acc0 = {}, acc1 = {}, acc2 = {}, acc3 = {};

#pragma unroll 4
    for (int k = 0; k < IN_DIM; k += 4) {
        v2f a  = *(const v2f*)(xa + k);
        v2f b0 = *(const v2f*)(w0 + k);
        v2f b1 = *(const v2f*)(w1 + k);
        v2f b2 = *(const v2f*)(w2 + k);
        v2f b3 = *(const v2f*)(w3 + k);
        acc0 = __builtin_amdgcn_wmma_f32_16x16x4_f32(false, a, false, b0,
                                                     (short)0, acc0, false, false);
        acc1 = __builtin_amdgcn_wmma_f32_16x16x4_f32(false, a, false, b1,
                                                     (short)0, acc1, false, false);
        acc2 = __builtin_amdgcn_wmma_f32_16x16x4_f32(false, a, false, b2,
                                                     (short)0, acc2, false, false);
        acc3 = __builtin_amdgcn_wmma_f32_16x16x4_f32(false, a, false, b3,
                                                     (short)0, acc3, false, false);
    }

    // C/D layout: VGPR i -> M = i + (lane<16 ? 0 : 8), N = (lane&15)
    const int mbase = m0 + ((lane >> 4) << 3);
    const int ncol  = lane & 15;
    const float bb0 = bias[n0 + ncol +  0];
    const float bb1 = bias[n0 + ncol + 16];
    const float bb2 = bias[n0 + ncol + 32];
    const float bb3 = bias[n0 + ncol + 48];

#pragma unroll
    for (int i = 0; i < 8; ++i) {
        float* hr = h + (size_t)(mbase + i) * OUT_DIM + n0 + ncol;
        hr[ 0] = acc0[i] + bb0;
        hr[16] = acc1[i] + bb1;
        hr[32] = acc2[i] + bb2;
        hr[48] = acc3[i] + bb3;
    }
}

// ---------------------------------------------------------------------------
// Kernel 2: out[dst[e]] += h[src[e]] * w_edge[e]
// One wave per edge: 2x float4 loads per lane (256 cols), 8 atomic f32 adds.
// h and out both fit in the 192MB L2, so this stage runs at L2 bandwidth.
// ---------------------------------------------------------------------------
__global__ void edge_scatter_kernel(const float* __restrict__ h,
                                    const float* __restrict__ w_edge,
                                    const int* __restrict__ src,
                                    const int* __restrict__ dst,
                                    float* __restrict__ out,
                                    int n_edges) {
    const int wave = threadIdx.x >> 5;
    const int lane = threadIdx.x & 31;
    const int e = blockIdx.x * (blockDim.x >> 5) + wave;
    if (e >= n_edges) return;                           // wave-uniform guard

    const int   s = src[e];
    const int   d = dst[e];
    const float w = w_edge[e];

    const float4* hs = (const float4*)(h + (size_t)s * OUT_DIM);
    float*        od = out + (size_t)d * OUT_DIM;

    float4 v0 = hs[lane];        // cols 4*lane   .. 4*lane+3
    float4 v1 = hs[lane + 32];   // cols 128+4*lane ..

    const int c0 = 4 * lane;
    atomicAdd(&od[c0 + 0],       v0.x * w);
    atomicAdd(&od[c0 + 1],       v0.y * w);
    atomicAdd(&od[c0 + 2],       v0.z * w);
    atomicAdd(&od[c0 + 3],       v0.w * w);
    atomicAdd(&od[c0 + 128 + 0], v1.x * w);
    atomicAdd(&od[c0 + 128 + 1], v1.y * w);
    atomicAdd(&od[c0 + 128 + 2], v1.z * w);
    atomicAdd(&od[c0 + 128 + 3], v1.w * w);
}

// ---------------------------------------------------------------------------
extern "C" void kernel_launch(void* const* d_in, const int* in_sizes, int n_in,
                              void* d_out, int out_size, void* d_ws, size_t ws_size,
                              hipStream_t stream) {
    const float* x      = (const float*)d_in[0];   // [N_NODES, IN_DIM]
    const float* W      = (const float*)d_in[1];   // [OUT_DIM, IN_DIM]
    const float* bias   = (const float*)d_in[2];   // [OUT_DIM]
    const float* w_edge = (const float*)d_in[3];   // [N_EDGES]
    const int*   src    = (const int*)d_in[4];     // [N_EDGES]
    const int*   dst    = (const int*)d_in[5];     // [N_EDGES]
    float*       out    = (float*)d_out;           // [N_NODES, OUT_DIM]
    float*       h      = (float*)d_ws;            // [N_NODES, OUT_DIM] scratch

    // 1) zero the segment-sum accumulator
    const int n4 = (N_NODES * OUT_DIM) / 4;
    zero_out_kernel<<<(n4 + 255) / 256, 256, 0, stream>>>((float4*)out, n4);

    // 2) h = x @ W^T + b   (WMMA f32, one 16x64 tile per wave, 8 waves/block)
    const int total_waves = (N_NODES / 16) * (OUT_DIM / 64);   // 3125 * 4
    gemm_wmma_f32_kernel<<<(total_waves + 7) / 8, 256, 0, stream>>>(x, W, bias, h);

    // 3) scatter-add over edges (one wave per edge, 8 waves/block)
    edge_scatter_kernel<<<(N_EDGES + 7) / 8, 256, 0, stream>>>(h, w_edge, src, dst,
                                                               out, N_EDGES);
}